// HAMLoss_86775519248543
// MI455X (gfx1250) — compile-verified
//
#include <hip/hip_runtime.h>
#include <hip/hip_bf16.h>
#include <math.h>

#define KSEL 5
#define T1F 0.35f
#define T2F 0.5f
#define ALPHAF 0.25f
#define BETAF 0.11f
#define VAR0F 0.1f
#define VAR1F 0.2f
#define MAXG 64
#define PPT 8
#define BLK 256

typedef unsigned long long u64;
typedef float v2f __attribute__((ext_vector_type(2)));
typedef float v8f __attribute__((ext_vector_type(8)));

// ---------- CDNA5 WMMA wave32 fp32 sum reduction ----------
// A (16x4 f32, 2 VGPRs/lane): A[m][0]=v[lane m], A[m][2]=v[lane m+16], odd-K = 0.
// B = all ones -> D[m][n] = v[m] + v[m+16] for every n (B layout irrelevant).
// C/D layout: VGPR r of lane L holds D[r + 8*(L>=16)][L&15]; summing the 8 D
// regs gives half-sums in each 16-lane half; one xor-16 shuffle completes it.
__device__ __forceinline__ float wave_sum_wmma(float v) {
  v2f a; a.x = v;   a.y = 0.0f;
  v2f b; b.x = 1.0f; b.y = 1.0f;
  v8f c = {};
  c = __builtin_amdgcn_wmma_f32_16x16x4_f32(false, a, false, b, (short)0, c,
                                            false, false);
  float s = c[0] + c[1] + c[2] + c[3] + c[4] + c[5] + c[6] + c[7];
  s += __shfl_xor(s, 16, 32);
  return s;
}

__device__ __forceinline__ float iou_box(float ax0, float ay0, float ax1, float ay1, float aA,
                                         float bx0, float by0, float bx1, float by1, float bA) {
  float ix0 = fmaxf(ax0, bx0), iy0 = fmaxf(ay0, by0);
  float ix1 = fminf(ax1, bx1), iy1 = fminf(ay1, by1);
  float w = fmaxf(ix1 - ix0, 0.f), h = fmaxf(iy1 - iy0, 0.f);
  float inter = w * h;
  return inter / (aA + bA - inter);
}

__device__ __forceinline__ float smooth_l1(float p, float t) {
  float x = fabsf(p - t);
  return (x >= BETAF) ? (x - 0.5f * BETAF) : (0.5f * x * x / BETAF);
}

__device__ __forceinline__ float focal_elem(float t, float x, float fiou) {
  float ce = fmaxf(x, 0.f) - x * t + log1pf(expf(-fabsf(x)));
  float a  = fiou * (t * ALPHAF + (1.f - t) * (1.f - ALPHAF));
  float pt = (t == 1.f) ? x : (1.f - x);
  float om = 1.f - pt;
  return a * om * om * ce;   // GAMMA == 2
}

__device__ __forceinline__ u64 block_max_u64(u64 k, u64* sm, int tid) {
  #pragma unroll
  for (int o = 16; o > 0; o >>= 1) {
    u64 other = __shfl_xor(k, o, 32);
    if (other > k) k = other;
  }
  if ((tid & 31) == 0) sm[tid >> 5] = k;
  __syncthreads();
  if (tid == 0) {
    int nw = (int)(blockDim.x >> 5);
    u64 m = sm[0];
    for (int i = 1; i < nw; i++) if (sm[i] > m) m = sm[i];
    sm[0] = m;
  }
  __syncthreads();
  u64 r = sm[0];
  __syncthreads();
  return r;
}

__global__ void ham_zero(int* a, int n) {
  int i = blockIdx.x * blockDim.x + threadIdx.x;
  if (i < n) a[i] = 0;
}

// Pass 1: matching, normal focal + smooth-L1 sums, match_cnt, npos.
__global__ void ham_pass1(const float* __restrict__ loc, const float* __restrict__ conf,
                          const float* __restrict__ priors, const float* __restrict__ targets,
                          int P, int G, int C,
                          int* __restrict__ match_cnt, int* __restrict__ nposA,
                          float* __restrict__ blk_fl, float* __restrict__ blk_sl) {
  __shared__ float s_t[MAXG][6];   // x0,y0,x1,y1,area,label
  __shared__ int   s_mc[MAXG];
  __shared__ int   s_np;
  __shared__ float s_wfl[8], s_wsl[8];
  int tid = threadIdx.x, b = blockIdx.y;
  if (tid < G) {
    const float* tg = targets + ((size_t)b * G + tid) * 5;
    float x0 = tg[0], y0 = tg[1], x1 = tg[2], y1 = tg[3];
    s_t[tid][0] = x0; s_t[tid][1] = y0; s_t[tid][2] = x1; s_t[tid][3] = y1;
    s_t[tid][4] = (x1 - x0) * (y1 - y0);
    s_t[tid][5] = tg[4];
    s_mc[tid] = 0;
  }
  if (tid == 0) s_np = 0;
  __syncthreads();

  size_t p = (size_t)blockIdx.x * blockDim.x + tid;
  const float* pr = priors + p * 4;
  float pcx = pr[0], pcy = pr[1], pw = pr[2], ph = pr[3];
  float fx0 = pcx - pw * 0.5f, fy0 = pcy - ph * 0.5f;
  float fx1 = pcx + pw * 0.5f, fy1 = pcy + ph * 0.5f;
  float fA = (fx1 - fx0) * (fy1 - fy0);
  const float* lp = loc + ((size_t)b * P + p) * 4;
  float l0 = lp[0], l1 = lp[1], l2 = lp[2], l3 = lp[3];
  float cx = pcx + l0 * VAR0F * pw, cy = pcy + l1 * VAR0F * ph;
  float w = pw * expf(l2 * VAR1F), h = ph * expf(l3 * VAR1F);
  float dx0 = cx - w * 0.5f, dy0 = cy - h * 0.5f, dx1 = cx + w * 0.5f, dy1 = cy + h * 0.5f;
  float dA = (dx1 - dx0) * (dy1 - dy0);
  const float* cf = conf + ((size_t)b * P + p) * C;
  __builtin_prefetch(cf, 0, 1);   // global_prefetch_b8

  float bs = -INFINITY; int bi = 0; float cmx = -INFINITY;
  for (int g = 0; g < G; g++) {
    float v = iou_box(s_t[g][0], s_t[g][1], s_t[g][2], s_t[g][3], s_t[g][4],
                      fx0, fy0, fx1, fy1, fA);
    if (v > bs) { bs = v; bi = g; }
    float cv = iou_box(s_t[g][0], s_t[g][1], s_t[g][2], s_t[g][3], s_t[g][4],
                       dx0, dy0, dx1, dy1, dA);
    if (cv > cmx) cmx = cv;
  }
  bool npos = bs > T1F;
  bool ign  = (cmx > T2F) && (bs < T1F);
  bool neg  = !(npos || ign);
  float fl = 0.f, sl = 0.f;
  if (npos) {
    int labi = (int)s_t[bi][5];
    for (int c = 0; c < C; c++) fl += focal_elem((c == labi) ? 1.f : 0.f, cf[c], 1.f);
    float tx0 = s_t[bi][0], ty0 = s_t[bi][1], tx1 = s_t[bi][2], ty1 = s_t[bi][3];
    float e0 = ((tx0 + tx1) * 0.5f - pcx) / (VAR0F * pw);
    float e1 = ((ty0 + ty1) * 0.5f - pcy) / (VAR0F * ph);
    float e2 = logf((tx1 - tx0) / pw) / VAR1F;
    float e3 = logf((ty1 - ty0) / ph) / VAR1F;
    sl = smooth_l1(l0, e0) + smooth_l1(l1, e1) + smooth_l1(l2, e2) + smooth_l1(l3, e3);
    atomicAdd(&s_mc[bi], 1);
    atomicAdd(&s_np, 1);
  } else if (neg) {
    for (int c = 0; c < C; c++) fl += focal_elem(0.f, cf[c], 1.f);
  }
  // uniform call site (EXEC all ones): WMMA-based fp32 wave reductions
  float flw = wave_sum_wmma(fl);
  float slw = wave_sum_wmma(sl);
  if ((tid & 31) == 0) { s_wfl[tid >> 5] = flw; s_wsl[tid >> 5] = slw; }
  __syncthreads();
  if (tid == 0) {
    int nw = (int)(blockDim.x >> 5);
    float tf = 0.f, ts = 0.f;
    for (int i = 0; i < nw; i++) { tf += s_wfl[i]; ts += s_wsl[i]; }
    blk_fl[(size_t)b * gridDim.x + blockIdx.x] = tf;
    blk_sl[(size_t)b * gridDim.x + blockIdx.x] = ts;
    if (s_np) atomicAdd(&nposA[b], s_np);
  }
  if (tid < G && s_mc[tid]) atomicAdd(&match_cnt[b * G + tid], s_mc[tid]);
}

// Pass 2: c_iou2 (with kill), masked argmax -> c_match, per-chunk top-5 keys.
__global__ void ham_pass2(const float* __restrict__ loc, const float* __restrict__ priors,
                          const float* __restrict__ targets, const int* __restrict__ match_cnt,
                          int* __restrict__ c_match, u64* __restrict__ top5chunk,
                          int P, int G) {
  __shared__ float s_t[MAXG][5];
  __shared__ int   s_outf[MAXG];
  __shared__ int   s_cm[MAXG];
  __shared__ u64   sm[8];
  int tid = threadIdx.x, b = blockIdx.y, chunk = blockIdx.x;
  int NCHUNK = gridDim.x;
  if (tid < G) {
    const float* tg = targets + ((size_t)b * G + tid) * 5;
    float x0 = tg[0], y0 = tg[1], x1 = tg[2], y1 = tg[3];
    s_t[tid][0] = x0; s_t[tid][1] = y0; s_t[tid][2] = x1; s_t[tid][3] = y1;
    s_t[tid][4] = (x1 - x0) * (y1 - y0);
    s_outf[tid] = (match_cnt[b * G + tid] < KSEL) ? 1 : 0;
    s_cm[tid] = 0;
  }
  __syncthreads();

  int base = chunk * (BLK * PPT);
  float dx0[PPT], dy0[PPT], dx1[PPT], dy1[PPT], dA[PPT];
  float bsc[PPT]; int bix[PPT];
  float cmv[PPT]; int cmi[PPT];
  #pragma unroll
  for (int i = 0; i < PPT; i++) {
    int p = base + tid + i * BLK;
    const float* pr = priors + (size_t)p * 4;
    float pcx = pr[0], pcy = pr[1], pw = pr[2], ph = pr[3];
    float fx0 = pcx - pw * 0.5f, fy0 = pcy - ph * 0.5f;
    float fx1 = pcx + pw * 0.5f, fy1 = pcy + ph * 0.5f;
    float fA = (fx1 - fx0) * (fy1 - fy0);
    const float* lp = loc + ((size_t)b * P + p) * 4;
    float cx = pcx + lp[0] * VAR0F * pw, cy = pcy + lp[1] * VAR0F * ph;
    float w = pw * expf(lp[2] * VAR1F), h = ph * expf(lp[3] * VAR1F);
    dx0[i] = cx - w * 0.5f; dy0[i] = cy - h * 0.5f;
    dx1[i] = cx + w * 0.5f; dy1[i] = cy + h * 0.5f;
    dA[i] = (dx1[i] - dx0[i]) * (dy1[i] - dy0[i]);
    float bs = -INFINITY; int bi = 0;
    for (int g = 0; g < G; g++) {
      float v = iou_box(s_t[g][0], s_t[g][1], s_t[g][2], s_t[g][3], s_t[g][4],
                        fx0, fy0, fx1, fy1, fA);
      if (v > bs) { bs = v; bi = g; }
    }
    bsc[i] = bs; bix[i] = bi;
    cmv[i] = -INFINITY; cmi[i] = 0;
  }

  for (int g = 0; g < G; g++) {
    float tx0 = s_t[g][0], ty0 = s_t[g][1], tx1 = s_t[g][2], ty1 = s_t[g][3], tA = s_t[g][4];
    int outf = s_outf[g];
    u64 kk[PPT];
    #pragma unroll
    for (int i = 0; i < PPT; i++) {
      float ci = iou_box(tx0, ty0, tx1, ty1, tA, dx0[i], dy0[i], dx1[i], dy1[i], dA[i]);
      if (outf && bix[i] == g && bsc[i] > T1F) ci = 0.f;   // kill
      float m = outf ? ci : -1.f;                           // masked
      if (m > cmv[i]) { cmv[i] = m; cmi[i] = g; }
      unsigned p = (unsigned)(base + tid + i * BLK);
      // nonneg f32 bits are order-preserving; ~index breaks ties toward low p
      kk[i] = ((u64)__float_as_uint(ci) << 32) | (u64)(0xFFFFFFFFu - p);
    }
    for (int it = 0; it < KSEL; it++) {
      u64 lk = 0;
      #pragma unroll
      for (int i = 0; i < PPT; i++) if (kk[i] > lk) lk = kk[i];
      u64 bm = block_max_u64(lk, sm, tid);
      #pragma unroll
      for (int i = 0; i < PPT; i++) if (kk[i] == bm) kk[i] = 0ull;
      if (tid == 0) top5chunk[((size_t)(b * G + g) * NCHUNK + chunk) * KSEL + it] = bm;
    }
  }
  #pragma unroll
  for (int i = 0; i < PPT; i++) if (cmv[i] > T2F) atomicAdd(&s_cm[cmi[i]], 1);
  __syncthreads();
  if (tid < G && s_cm[tid]) atomicAdd(&c_match[b * G + tid], s_cm[tid]);
}

// Pass 3: merge chunk top-5 -> row top-5 (descending), c_anchor_num, csum.
__global__ void ham_pass3(const u64* __restrict__ top5chunk, u64* __restrict__ top5,
                          const int* __restrict__ match_cnt, const int* __restrict__ c_match,
                          int* __restrict__ c_anchor, int* __restrict__ csum,
                          int G, int NCHUNK) {
  __shared__ u64 sm[8];
  int tid = threadIdx.x, g = blockIdx.x, b = blockIdx.y;
  int NC = NCHUNK * KSEL;     // 160 <= blockDim for this problem size
  u64 k = 0;
  if (tid < NC) k = top5chunk[(size_t)(b * G + g) * NC + tid];
  for (int it = 0; it < KSEL; it++) {
    u64 bm = block_max_u64(k, sm, tid);
    if (k == bm) k = 0;
    if (tid == 0) top5[(size_t)(b * G + g) * KSEL + it] = bm;
  }
  if (tid == 0) {
    int mc = match_cnt[b * G + g], cm = c_match[b * G + g];
    int n = (mc < KSEL) ? ((cm < KSEL - mc) ? cm : (KSEL - mc)) : 0;
    c_anchor[b * G + g] = n;
    if (n) atomicAdd(&csum[b], n);
  }
}

// Pass 4: sequential g-scan over selected anchors, finalize per-image losses.
__global__ void ham_pass4(const float* __restrict__ loc, const float* __restrict__ conf,
                          const float* __restrict__ priors, const float* __restrict__ targets,
                          const u64* __restrict__ top5, const int* __restrict__ c_anchor,
                          const int* __restrict__ nposA, const int* __restrict__ csum,
                          const float* __restrict__ blk_fl, const float* __restrict__ blk_sl,
                          float* __restrict__ img_loc, float* __restrict__ img_cls,
                          int P, int G, int C, int nblk) {
  __shared__ int   s_p[MAXG * KSEL];
  __shared__ int   s_gi[MAXG * KSEL];
  __shared__ float s_iou[MAXG * KSEL];
  __shared__ float s_sle[MAXG * KSEL];
  __shared__ float s_fle[MAXG * KSEL];
  __shared__ int   s_cnt;
  int tid = threadIdx.x, b = blockIdx.x;
  if (tid == 0) {
    int cnt = 0;
    for (int g = 0; g < G; g++) {           // ascending g => later overwrites (scan)
      int n = c_anchor[b * G + g];
      for (int j = 0; j < n; j++) {
        u64 key = top5[(size_t)(b * G + g) * KSEL + j];
        int pp = (int)(0xFFFFFFFFu - (unsigned)(key & 0xFFFFFFFFull));
        float val = __uint_as_float((unsigned)(key >> 32));
        int f = -1;
        for (int q = 0; q < cnt; q++) if (s_p[q] == pp) { f = q; break; }
        if (f < 0) { f = cnt++; s_p[f] = pp; }
        s_gi[f] = g; s_iou[f] = val;
      }
    }
    s_cnt = cnt;
  }
  __syncthreads();
  int cnt = s_cnt;
  for (int t = tid; t < cnt; t += blockDim.x) {
    int p = s_p[t]; int g = s_gi[t]; float fi = s_iou[t];
    const float* tg = targets + ((size_t)b * G + g) * 5;
    float tx0 = tg[0], ty0 = tg[1], tx1 = tg[2], ty1 = tg[3], lab = tg[4];
    const float* pr = priors + (size_t)p * 4;
    float pcx = pr[0], pcy = pr[1], pw = pr[2], ph = pr[3];
    const float* lp = loc + ((size_t)b * P + p) * 4;
    float e0 = ((tx0 + tx1) * 0.5f - pcx) / (VAR0F * pw);
    float e1 = ((ty0 + ty1) * 0.5f - pcy) / (VAR0F * ph);
    float e2 = logf((tx1 - tx0) / pw) / VAR1F;
    float e3 = logf((ty1 - ty0) / ph) / VAR1F;
    s_sle[t] = smooth_l1(lp[0], e0) + smooth_l1(lp[1], e1) +
               smooth_l1(lp[2], e2) + smooth_l1(lp[3], e3);
    const float* cf = conf + ((size_t)b * P + p) * C;
    int labi = (int)lab;
    float flv = 0.f;
    for (int c = 0; c < C; c++) {
      float tv = (c == labi) ? lab : 0.f;   // labels[g] * one_hot
      flv += focal_elem(tv, cf[c], fi);
    }
    s_fle[t] = flv;
  }
  __syncthreads();
  if (tid == 0) {
    float slc = 0.f, flc = 0.f;
    for (int t = 0; t < cnt; t++) { slc += s_sle[t]; flc += s_fle[t]; }
    float fsum = 0.f, ssum = 0.f;            // deterministic fixed-order sum
    for (int i = 0; i < nblk; i++) {
      fsum += blk_fl[(size_t)b * nblk + i];
      ssum += blk_sl[(size_t)b * nblk + i];
    }
    int np = nposA[b];
    float npf = (float)((np > 1) ? np : 1);
    float lln = (np > 0) ? ssum / npf : 0.f;
    float lcn = (np > 0) ? fsum / npf : 0.f;
    int cs = csum[b];
    int npc = (cnt > 1) ? cnt : 1;
    int csm = (cs > 1) ? cs : 1;
    float llc = (cs > 0) ? slc / (float)npc : 0.f;
    float lcc = (cs > 0) ? flc / (float)csm : 0.f;
    img_loc[b] = lln + llc;
    img_cls[b] = lcn + lcc;
  }
}

__global__ void ham_final(const float* img_loc, const float* img_cls, float* out, int B) {
  if (threadIdx.x == 0 && blockIdx.x == 0) {
    float a = 0.f, c = 0.f;
    for (int b = 0; b < B; b++) { a += img_loc[b]; c += img_cls[b]; }
    out[0] = a / (float)B;    // loss_loc mean
    out[1] = c / (float)B;    // loss_cls mean
  }
}

extern "C" void kernel_launch(void* const* d_in, const int* in_sizes, int n_in,
                              void* d_out, int out_size, void* d_ws, size_t ws_size,
                              hipStream_t stream) {
  (void)n_in; (void)out_size; (void)ws_size;
  const float* loc     = (const float*)d_in[0];
  const float* conf    = (const float*)d_in[1];
  const float* priors  = (const float*)d_in[2];
  const float* targets = (const float*)d_in[3];
  int P = in_sizes[2] / 4;
  int B = in_sizes[0] / (P * 4);
  int C = in_sizes[1] / (B * P);
  int G = in_sizes[3] / (B * 5);
  int nblk   = P / BLK;            // 256
  int NCHUNK = P / (BLK * PPT);    // 32

  char* ws = (char*)d_ws;
  size_t o = 0;
  auto carve = [&](size_t bytes) -> char* {
    char* q = ws + o; o = (o + bytes + 15) & ~(size_t)15; return q;
  };
  u64*   top5chunk = (u64*)  carve((size_t)B * G * NCHUNK * KSEL * 8);
  u64*   top5      = (u64*)  carve((size_t)B * G * KSEL * 8);
  float* blk_fl    = (float*)carve((size_t)B * nblk * 4);
  float* blk_sl    = (float*)carve((size_t)B * nblk * 4);
  size_t zbeg = o;
  int*   match_cnt = (int*)  carve((size_t)B * G * 4);
  int*   c_match   = (int*)  carve((size_t)B * G * 4);
  int*   nposA     = (int*)  carve((size_t)B * 4);
  int*   csum      = (int*)  carve((size_t)B * 4);
  size_t zend = o;
  int*   c_anchor  = (int*)  carve((size_t)B * G * 4);
  float* img_loc   = (float*)carve((size_t)B * 4);
  float* img_cls   = (float*)carve((size_t)B * 4);

  int nz = (int)((zend - zbeg) / 4);
  ham_zero<<<(nz + 255) / 256, 256, 0, stream>>>((int*)(ws + zbeg), nz);
  ham_pass1<<<dim3(nblk, B), BLK, 0, stream>>>(loc, conf, priors, targets, P, G, C,
                                               match_cnt, nposA, blk_fl, blk_sl);
  ham_pass2<<<dim3(NCHUNK, B), BLK, 0, stream>>>(loc, priors, targets, match_cnt,
                                                 c_match, top5chunk, P, G);
  ham_pass3<<<dim3(G, B), BLK, 0, stream>>>(top5chunk, top5, match_cnt, c_match,
                                            c_anchor, csum, G, NCHUNK);
  ham_pass4<<<dim3(B), BLK, 0, stream>>>(loc, conf, priors, targets, top5, c_anchor,
                                         nposA, csum, blk_fl, blk_sl,
                                         img_loc, img_cls, P, G, C, nblk);
  ham_final<<<1, 32, 0, stream>>>(img_loc, img_cls, (float*)d_out, B);
}